// OuterProductMean_16716012716469
// MI455X (gfx1250) — compile-verified
//
#include <hip/hip_runtime.h>

typedef __attribute__((ext_vector_type(2))) float v2f;
typedef __attribute__((ext_vector_type(4))) float v4f;
typedef __attribute__((ext_vector_type(8))) float v8f;

#define N_TOK 512
#define C_M   256
#define C_H   32
#define C_Z   128
#define LN_EPS 1e-5f
#define TPAD  36   // padded LDS row stride (floats): 16B-aligned float4, bank-conflict-free

// ---------------------------------------------------------------------------
// Kernel 1: LayerNorm over c_m + dual projections -> a[i,h], b[i,h]
// one block per token row i, 256 threads (one per c_m element)
// ---------------------------------------------------------------------------
__global__ __launch_bounds__(256) void k_ln_proj(
    const float* __restrict__ m,
    const float* __restrict__ gamma, const float* __restrict__ beta,
    const float* __restrict__ w1, const float* __restrict__ b1,
    const float* __restrict__ w2, const float* __restrict__ b2,
    float* __restrict__ a_rows, float* __restrict__ b_rows)
{
    __shared__ float red[256];
    __shared__ float lnrow[C_M];
    __shared__ float part[64][4];

    const int i   = blockIdx.x;
    const int tid = threadIdx.x;

    const float x = m[i * C_M + tid];

    // mean
    red[tid] = x; __syncthreads();
    #pragma unroll
    for (int s = 128; s > 0; s >>= 1) {
        if (tid < s) red[tid] += red[tid + s];
        __syncthreads();
    }
    const float mean = red[0] * (1.0f / C_M);
    __syncthreads();

    // variance (population)
    red[tid] = x * x; __syncthreads();
    #pragma unroll
    for (int s = 128; s > 0; s >>= 1) {
        if (tid < s) red[tid] += red[tid + s];
        __syncthreads();
    }
    const float var = red[0] * (1.0f / C_M) - mean * mean;
    const float rs  = rsqrtf(var + LN_EPS);

    lnrow[tid] = (x - mean) * rs * gamma[tid] + beta[tid];
    __syncthreads();

    // 64 outputs (a: h=0..31, b: h=0..31), 4 partial sums each
    {
        const int o   = tid & 63;          // output id
        const int q   = tid >> 6;          // quarter of the c_m axis
        const int h   = o & 31;
        const int sel = o >> 5;            // 0 -> w1/a, 1 -> w2/b
        const float* __restrict__ w = sel ? w2 : w1;
        float s = 0.0f;
        const int base = q * 64;
        #pragma unroll 8
        for (int c = 0; c < 64; ++c)
            s += lnrow[base + c] * w[h * C_M + base + c];
        part[o][q] = s;
    }
    __syncthreads();

    if (tid < 64) {
        const int h   = tid & 31;
        const int sel = tid >> 5;
        float r = part[tid][0] + part[tid][1] + part[tid][2] + part[tid][3];
        r += sel ? b2[h] : b1[h];
        (sel ? b_rows : a_rows)[i * C_H + h] = r;
    }
}

// ---------------------------------------------------------------------------
// Kernel 2: t[i,z,d] = sum_c a[i,c] * w_out[z*1024 + c*32 + d]
// one block per token i, 256 threads, 16 outputs/thread (134 MFLOP total)
// ---------------------------------------------------------------------------
__global__ __launch_bounds__(256) void k_mid(
    const float* __restrict__ a_rows,
    const float* __restrict__ w_out,
    float* __restrict__ tmat)
{
    __shared__ float arow[C_H];
    const int i   = blockIdx.x;
    const int tid = threadIdx.x;
    if (tid < C_H) arow[tid] = a_rows[i * C_H + tid];
    __syncthreads();

    for (int e = tid; e < C_Z * C_H; e += 256) {
        const int z = e >> 5;
        const int d = e & 31;
        float s = 0.0f;
        #pragma unroll
        for (int c = 0; c < C_H; ++c)
            s += arow[c] * w_out[z * (C_H * C_H) + c * C_H + d];
        tmat[i * (C_Z * C_H) + e] = s;
    }
}

// ---------------------------------------------------------------------------
// Kernel 3 (the hot one): out[i,j,z] = sum_d t[i,z,d]*b[j,d] + b_out[z]
// Per i: Out_i[512 x 128] = B[512 x 32] x T_i^T[32 x 128], fp32 WMMA 16x16x4.
// grid = (4 j-blocks, 512 i); block = 256 threads = 8 waves.
// Wave w computes a 16-j strip x 128 z = 8 accumulator tiles.
// ---------------------------------------------------------------------------
__global__ __launch_bounds__(256) void k_outer(
    const float* __restrict__ tmat,    // [512][128][32]
    const float* __restrict__ b_rows,  // [512][32]
    const float* __restrict__ bias,    // [128]
    float* __restrict__ out)           // [512][512][128]
{
    __shared__ float lds_t[C_Z * TPAD];   // t_i, padded rows (z-major)
    __shared__ float lds_b[128 * TPAD];   // 128 b rows for this j-block

    const int jb  = blockIdx.x;           // j-block (128 rows)
    const int i   = blockIdx.y;           // token
    const int tid = threadIdx.x;

    // ---- stage t_i and b-slice into LDS (float4, 16B aligned at stride 36) --
    const float* __restrict__ tsrc = tmat + (size_t)i * (C_Z * C_H);
    const float* __restrict__ bsrc = b_rows + (size_t)jb * 128 * C_H;
    for (int e4 = tid; e4 < (C_Z * C_H) / 4; e4 += 256) {
        const int row = e4 >> 3;           // /8 float4 per 32-float row
        const int d4  = (e4 & 7) << 2;
        *(v4f*)&lds_t[row * TPAD + d4] = *(const v4f*)(tsrc + row * C_H + d4);
        *(v4f*)&lds_b[row * TPAD + d4] = *(const v4f*)(bsrc + row * C_H + d4);
    }
    __syncthreads();

    const int lane = tid & 31;
    const int wav  = tid >> 5;
    const int l15  = lane & 15;
    const int half = lane >> 4;            // lane half selects K pair / M half

    // ---- accumulators: 8 z-tiles, bias folded into init ---------------------
    v8f acc[8];
    #pragma unroll
    for (int zt = 0; zt < 8; ++zt) {
        const float bo = bias[zt * 16 + l15];   // C layout: N(=z) = lane%16
        #pragma unroll
        for (int r = 0; r < 8; ++r) acc[zt][r] = bo;
    }

    // ---- A fragments (b rows, 16x4 per K-step), hoisted & reused ------------
    // A layout: lane = M row (both halves), VGPR pair = K {0,1} / {2,3} by half
    v2f af[8];
    const int arow = wav * 16 + l15;
    #pragma unroll
    for (int k = 0; k < 8; ++k)
        af[k] = *(const v2f*)&lds_b[arow * TPAD + 4 * k + 2 * half];

    // ---- main WMMA loop: K-step outer, z-tile inner (8 indep. accumulators) -
    #pragma unroll
    for (int k = 0; k < 8; ++k) {
        #pragma unroll
        for (int zt = 0; zt < 8; ++zt) {
            // B layout (4x16): lane = N(=z), VGPR pair = K {0,1}/{2,3} by half
            const int z = zt * 16 + l15;
            const v2f bf = *(const v2f*)&lds_t[z * TPAD + 4 * k + 2 * half];
            acc[zt] = __builtin_amdgcn_wmma_f32_16x16x4_f32(
                false, af[k], false, bf, (short)0, acc[zt], false, false);
        }
    }

    // ---- store: C layout lane<16 -> M=r, lane>=16 -> M=8+r; N = lane%16 -----
    float* __restrict__ obase =
        out + (size_t)i * N_TOK * C_Z + (size_t)(jb * 128 + wav * 16) * C_Z;
    #pragma unroll
    for (int zt = 0; zt < 8; ++zt) {
        #pragma unroll
        for (int r = 0; r < 8; ++r) {
            const int jrow = half ? (8 + r) : r;
            obase[jrow * C_Z + zt * 16 + l15] = acc[zt][r];
        }
    }
}

// ---------------------------------------------------------------------------
extern "C" void kernel_launch(void* const* d_in, const int* in_sizes, int n_in,
                              void* d_out, int out_size, void* d_ws, size_t ws_size,
                              hipStream_t stream)
{
    const float* m      = (const float*)d_in[0];
    const float* gamma  = (const float*)d_in[1];
    const float* beta   = (const float*)d_in[2];
    const float* w1     = (const float*)d_in[3];
    const float* b1     = (const float*)d_in[4];
    const float* w2     = (const float*)d_in[5];
    const float* b2     = (const float*)d_in[6];
    const float* w_out  = (const float*)d_in[7];
    const float* b_out  = (const float*)d_in[8];
    float* out = (float*)d_out;

    // workspace layout
    char* ws = (char*)d_ws;
    float* a_rows = (float*)ws;                                  // 512*32
    float* b_rows = (float*)(ws + 64 * 1024);                    // 512*32
    float* tmat   = (float*)(ws + 128 * 1024);                   // 512*128*32

    k_ln_proj<<<N_TOK, 256, 0, stream>>>(m, gamma, beta, w1, b1, w2, b2,
                                         a_rows, b_rows);
    k_mid<<<N_TOK, 256, 0, stream>>>(a_rows, w_out, tmat);
    k_outer<<<dim3(4, N_TOK), 256, 0, stream>>>(tmat, b_rows, b_out, out);
}